// DES_PSP_Model_90005334654993
// MI455X (gfx1250) — compile-verified
//
#include <hip/hip_runtime.h>
#include <hip/hip_bf16.h>

// MI455X / gfx1250. f32 WMMA (16x16x4) for all matrix math; TDM async-tensor
// staging for conv inputs; persistent barrier-free LSTM workgroups; hardware
// TRANS ops (v_tanh_f32 / v_rcp_f32 / v_exp_f32) for the gate nonlinearities.

#define B_   128
#define T_   256
#define HID  64
#define G_   256   // 4*HID
#define L_   5
#define PS_  14
#define ALPHA_ 0.2f

typedef float v2f __attribute__((ext_vector_type(2)));
typedef float v8f __attribute__((ext_vector_type(8)));
typedef unsigned int u32x4 __attribute__((ext_vector_type(4)));
typedef int i32x4 __attribute__((ext_vector_type(4)));
typedef int i32x8 __attribute__((ext_vector_type(8)));

__device__ __forceinline__ v8f wmma_f32(v2f a, v2f b, v8f c) {
  return __builtin_amdgcn_wmma_f32_16x16x4_f32(false, a, false, b, (short)0, c,
                                               false, false);
}

__device__ __forceinline__ float fast_tanh(float x) {
#if __has_builtin(__builtin_amdgcn_tanhf)
  return __builtin_amdgcn_tanhf(x);            // V_TANH_F32 (gfx1250 TRANS op)
#elif __has_builtin(__builtin_amdgcn_tanh_f32)
  return __builtin_amdgcn_tanh_f32(x);
#else
  const float e = __expf(-2.0f * x);
  return (1.0f - e) * __builtin_amdgcn_rcpf(1.0f + e);
#endif
}

__device__ __forceinline__ float sigmf(float x) {
  // v_exp_f32 + v_rcp_f32, no precise-division expansion
  return __builtin_amdgcn_rcpf(1.0f + __expf(-x));
}

// ---------------------------------------------------------------------------
// TDM: 1-D tile copy global -> LDS (nelem f32 elements). D# per ISA 8.3/8.4:
// group0 {count=1, lds_addr, global_addr, type=2}; group1 {data_size=4B,
// tensor_dim0=nelem, dim1=1, tile_dim0=nelem, tile_dim1=1, stride0=nelem}.
// Issued from a single wave (TDM ignores EXEC); wait with s_wait_tensorcnt.
// ---------------------------------------------------------------------------
__device__ __forceinline__ void tdm_load_1d(const void* gptr, unsigned lds_off,
                                            unsigned nelem) {
  unsigned long long ga = (unsigned long long)(size_t)gptr;
  u32x4 g0;
  g0[0] = 1u;                                   // count=1 (valid), user mode
  g0[1] = lds_off;                              // LDS byte address
  g0[2] = (unsigned)(ga & 0xFFFFFFFFu);         // global_addr[31:0]
  g0[3] = (unsigned)((ga >> 32) & 0x1FFFFFFu)   // global_addr[56:32]
          | (2u << 30);                         // type=2 ("image")
  i32x8 g1;
  g1[0] = (2 << 16);                            // data_size=4B, mask=0
  g1[1] = (int)((nelem & 0xFFFFu) << 16);       // tensor_dim0[15:0] @63:48
  g1[2] = (int)(((nelem >> 16) & 0xFFFFu)       // tensor_dim0[31:16] @79:64
                | (1u << 16));                  // tensor_dim1=1 (low16) @95:80
  g1[3] = (int)((nelem & 0xFFFFu) << 16);       // tile_dim0 @127:112
  g1[4] = 1;                                    // tile_dim1=1, tile_dim2=0
  g1[5] = (int)nelem;                           // tensor_dim0_stride[31:0]
  g1[6] = 0;
  g1[7] = 0;
  i32x4 gz = (i32x4)0;
#if __clang_major__ >= 23
  i32x8 gz8 = (i32x8)0;
  __builtin_amdgcn_tensor_load_to_lds(g0, g1, gz, gz, gz8, 0);
#else
  __builtin_amdgcn_tensor_load_to_lds(g0, g1, gz, gz, 0);
#endif
  __builtin_amdgcn_s_wait_tensorcnt(0);
}

// ---------------------------------------------------------------------------
// Fused LSTM layer (encoder): grid=1, block=256 (8 wave32). Wave w owns batch
// rows 16w..16w+15 -> h/c rows in LDS are wave-private, zero barriers over the
// 256 sequential steps. IN_DIM templated so all guards are compile-time.
// ---------------------------------------------------------------------------
template <int IN_DIM, int TSTRIDE, int BSTRIDE>
__global__ __launch_bounds__(256) void lstm_layer_k(
    const float* __restrict__ seq_in,
    const float* __restrict__ Wih,   // [G_, IN_DIM]
    const float* __restrict__ Whh,   // [G_, HID]
    const float* __restrict__ bias,  // [G_]
    float* __restrict__ seq_out,     // [T_, B_, HID]
    float* __restrict__ h_out, float* __restrict__ c_out)
{
  __shared__ float hS[B_ * HID];  // 32 KB
  __shared__ float cS[B_ * HID];  // 32 KB

  const int lane = threadIdx.x & 31;
  const int wave = threadIdx.x >> 5;
  const int half = lane >> 4;
  const int r    = lane & 15;
  const int m0   = wave << 4;

  for (int i = lane; i < 16 * HID; i += 32) {
    hS[(m0 + (i >> 6)) * HID + (i & 63)] = 0.0f;
    cS[(m0 + (i >> 6)) * HID + (i & 63)] = 0.0f;
  }

  float bfrag[4][4];
#pragma unroll
  for (int g = 0; g < 4; ++g)
#pragma unroll
    for (int ng = 0; ng < 4; ++ng)
      bfrag[g][ng] = bias[g * HID + ng * 16 + r];

  for (int t = 0; t < T_; ++t) {
    float hnew[4][8];
#pragma unroll
    for (int ng = 0; ng < 4; ++ng) {
      v8f acc[4];
#pragma unroll
      for (int g = 0; g < 4; ++g) {
        const float bv = bfrag[g][ng];
#pragma unroll
        for (int j = 0; j < 8; ++j) acc[g][j] = bv;
      }

      // ---- x-projection ----
      if (IN_DIM == 1) {
        // K=1 padded to 4: unconditional loads + lane select (no branches)
        const float xv = seq_in[(size_t)t * TSTRIDE + (size_t)(m0 + r) * BSTRIDE];
        v2f a;
        a.x = half ? 0.0f : xv;
        a.y = 0.0f;
#pragma unroll
        for (int g = 0; g < 4; ++g) {
          const int n = g * HID + ng * 16 + r;
          const float wv = Wih[n];
          v2f b;
          b.x = half ? 0.0f : wv;
          b.y = 0.0f;
          acc[g] = wmma_f32(a, b, acc[g]);
        }
      } else {
#pragma unroll 4
        for (int kk = 0; kk < IN_DIM / 4; ++kk) {
          const int k0 = (kk << 2) + (half << 1);
          const v2f a = *(const v2f*)&seq_in[(size_t)t * TSTRIDE +
                                             (size_t)(m0 + r) * BSTRIDE + k0];
#pragma unroll
          for (int g = 0; g < 4; ++g) {
            const int n = g * HID + ng * 16 + r;
            const v2f b = *(const v2f*)&Wih[(size_t)n * IN_DIM + k0];
            acc[g] = wmma_f32(a, b, acc[g]);
          }
        }
      }

      // ---- recurrent projection: A from LDS (b64), B from global (b64) ----
#pragma unroll 4
      for (int kk = 0; kk < 16; ++kk) {
        const int k0 = (kk << 2) + (half << 1);
        const v2f a = *(const v2f*)&hS[(m0 + r) * HID + k0];
#pragma unroll
        for (int g = 0; g < 4; ++g) {
          const int n = g * HID + ng * 16 + r;
          const v2f b = *(const v2f*)&Whh[(size_t)n * HID + k0];
          acc[g] = wmma_f32(a, b, acc[g]);
        }
      }

      // ---- gate math: i/f/g/o for one (b,n) are in the same lane ----
      const int n = ng * 16 + r;
#pragma unroll
      for (int j = 0; j < 8; ++j) {
        const int row = m0 + j + (half << 3);
        const float iv = sigmf(acc[0][j]);
        const float fv = sigmf(acc[1][j]);
        const float gv = fast_tanh(acc[2][j]);
        const float ov = sigmf(acc[3][j]);
        const float cv = fv * cS[row * HID + n] + iv * gv;
        cS[row * HID + n] = cv;
        const float hv = ov * fast_tanh(cv);
        hnew[ng][j] = hv;
        seq_out[((size_t)t * B_ + row) * HID + n] = hv;
      }
    }
#pragma unroll
    for (int ng = 0; ng < 4; ++ng) {
      const int n = ng * 16 + r;
#pragma unroll
      for (int j = 0; j < 8; ++j)
        hS[(m0 + j + (half << 3)) * HID + n] = hnew[ng][j];
    }
  }

  for (int i = lane; i < 16 * HID; i += 32) {
    const int row = m0 + (i >> 6), col = i & 63;
    h_out[row * HID + col] = hS[row * HID + col];
    c_out[row * HID + col] = cS[row * HID + col];
  }
}

// ---------------------------------------------------------------------------
// conv0 (1->64ch, 32x32) fused with 2x2 mean-pool -> NHWC [B][256 pix][64 c]
// ---------------------------------------------------------------------------
__global__ __launch_bounds__(256) void conv0_pool_k(
    const float* __restrict__ y, const float* __restrict__ W0,
    const float* __restrict__ b0, float* __restrict__ z)
{
  const int idx = blockIdx.x * 256 + threadIdx.x;
  const int c = idx & 63, px = (idx >> 6) & 15, py = (idx >> 10) & 15;
  const int n = idx >> 14;
  const float* yi = y + (size_t)n * 1024;
  float s = 0.0f;
#pragma unroll
  for (int dy = 0; dy < 2; ++dy)
#pragma unroll
    for (int dx = 0; dx < 2; ++dx) {
      const int Y = 2 * py + dy, X = 2 * px + dx;
#pragma unroll
      for (int ky = 0; ky < 3; ++ky)
#pragma unroll
        for (int kx = 0; kx < 3; ++kx) {
          const int iy = Y + ky - 1, ix = X + kx - 1;
          const bool ok = (iy >= 0) & (iy < 32) & (ix >= 0) & (ix < 32);
          const int ia = ok ? (iy * 32 + ix) : 0;
          const float v = yi[ia];
          s += (ok ? v : 0.0f) * W0[c * 9 + ky * 3 + kx];
        }
    }
  z[((size_t)n * 256 + py * 16 + px) * 64 + c] = 0.25f * s + b0[c];
}

// repack conv weights [7][oc][ic][3][3] -> [7][oc][kpos][ic] for contiguous B
__global__ void repack_convW_k(const float* __restrict__ W,
                               float* __restrict__ Wr) {
  const int idx = blockIdx.x * 256 + threadIdx.x;  // 7*64*9*64 total
  const int ic = idx & 63, kpos = (idx >> 6) % 9;
  const int oc = (idx / 576) & 63, l = idx / (576 * 64);
  Wr[idx] = W[(((size_t)l * 64 + oc) * 64 + ic) * 9 + kpos];
}

// ---------------------------------------------------------------------------
// 3x3 conv 64->64 on 16x16 + ReLU, implicit GEMM via WMMA, NHWC activations.
// Input tile staged to LDS with the Tensor Data Mover (async tensor load).
// Out = [pix=256] x [oc=64]: 16 M-tiles x 4 N-tiles, K = 9 x 64.
// ---------------------------------------------------------------------------
__global__ __launch_bounds__(256) void conv3x3_relu_k(
    const float* __restrict__ in,   // [B][256][64] NHWC
    const float* __restrict__ Wr,   // [64 oc][9 kpos][64 ic]
    const float* __restrict__ bias, // [64]
    float* __restrict__ outp)       // [B][256][64] NHWC
{
  __shared__ float sIn[256 * 64];  // 64 KB
  const int n = blockIdx.x;
  const size_t ibase = (size_t)n * 256 * 64;

  if (threadIdx.x < 32) {  // one wave issues the TDM DMA and waits on it
    tdm_load_1d(in + ibase, (unsigned)(size_t)sIn, 256 * 64);
  }
  __builtin_prefetch(&Wr[threadIdx.x * 64], 0, 1);
  __syncthreads();

  const int lane = threadIdx.x & 31;
  const int wave = threadIdx.x >> 5;
  const int half = lane >> 4;
  const int r    = lane & 15;
  const int nt   = wave & 3;
  const int oc   = nt * 16 + r;
  const float bv = bias[oc];

  for (int mi = 0; mi < 8; ++mi) {
    const int mt = (wave >> 2) + (mi << 1);  // M-tile == image row mt
    v8f acc;
#pragma unroll
    for (int j = 0; j < 8; ++j) acc[j] = bv;

#pragma unroll
    for (int ky = 0; ky < 3; ++ky)
#pragma unroll
      for (int kx = 0; kx < 3; ++kx) {
        const int iy = mt + ky - 1;
        const int ix = r + kx - 1;
        const bool ok = (iy >= 0) & (iy < 16) & (ix >= 0) & (ix < 16);
        const int pix = ok ? (iy * 16 + ix) : 0;  // clamped addr + select
        const int kb = (ky * 3 + kx) * 64;
#pragma unroll 4
        for (int kk = 0; kk < 16; ++kk) {
          const int ic0 = (kk << 2) + (half << 1);
          const v2f av = *(const v2f*)&sIn[pix * 64 + ic0];
          v2f a;
          a.x = ok ? av.x : 0.0f;
          a.y = ok ? av.y : 0.0f;
          const v2f b = *(const v2f*)&Wr[oc * 576 + kb + ic0];
          acc = wmma_f32(a, b, acc);
        }
      }
#pragma unroll
    for (int j = 0; j < 8; ++j) {
      const int pix = mt * 16 + j + (half << 3);
      outp[ibase + (size_t)pix * 64 + oc] = fmaxf(acc[j], 0.0f);
    }
  }
}

// feat[n][c] = mean over 256 pixels (NHWC input)
__global__ void feat_mean_k(const float* __restrict__ z, float* __restrict__ feat) {
  const int n = blockIdx.x, c = threadIdx.x;
  const float* p = z + (size_t)n * 256 * 64 + c;
  float s = 0.0f;
  for (int i = 0; i < 256; ++i) s += p[i * 64];
  feat[n * 64 + c] = s * (1.0f / 256.0f);
}

// ---------------------------------------------------------------------------
// Decoder LSTM cell, IN_DIM templated (1: scalar input from inp[], 64: from
// lin[]). State rows wave-private; commits h to state + lin after all reads.
// ---------------------------------------------------------------------------
template <int IN_DIM>
__device__ __forceinline__ void dec_cell(
    const float* __restrict__ Wih, const float* __restrict__ Wh,
    const float* __restrict__ bias,
    float* hl, float* cl,          // global state [B_*HID] (no restrict: RMW)
    float* lin, const float* inp,  // LDS buffers
    int m0, int half, int r)
{
  float hnew[4][8];
#pragma unroll
  for (int ng = 0; ng < 4; ++ng) {
    v8f acc[4];
#pragma unroll
    for (int g = 0; g < 4; ++g) {
      const float bv = bias[g * HID + ng * 16 + r];
#pragma unroll
      for (int j = 0; j < 8; ++j) acc[g][j] = bv;
    }
    if (IN_DIM == 1) {
      const float xv = inp[m0 + r];
      v2f a;
      a.x = half ? 0.0f : xv;
      a.y = 0.0f;
#pragma unroll
      for (int g = 0; g < 4; ++g) {
        const int n = g * HID + ng * 16 + r;
        const float wv = Wih[n];
        v2f b;
        b.x = half ? 0.0f : wv;
        b.y = 0.0f;
        acc[g] = wmma_f32(a, b, acc[g]);
      }
    } else {
#pragma unroll 4
      for (int kk = 0; kk < 16; ++kk) {
        const int k0 = (kk << 2) + (half << 1);
        const v2f a = *(const v2f*)&lin[(m0 + r) * HID + k0];
#pragma unroll
        for (int g = 0; g < 4; ++g) {
          const int n = g * HID + ng * 16 + r;
          const v2f b = *(const v2f*)&Wih[(size_t)n * HID + k0];
          acc[g] = wmma_f32(a, b, acc[g]);
        }
      }
    }
#pragma unroll 4
    for (int kk = 0; kk < 16; ++kk) {
      const int k0 = (kk << 2) + (half << 1);
      const v2f a = *(const v2f*)&hl[(m0 + r) * HID + k0];
#pragma unroll
      for (int g = 0; g < 4; ++g) {
        const int n = g * HID + ng * 16 + r;
        const v2f b = *(const v2f*)&Wh[(size_t)n * HID + k0];
        acc[g] = wmma_f32(a, b, acc[g]);
      }
    }
    const int n = ng * 16 + r;
#pragma unroll
    for (int j = 0; j < 8; ++j) {
      const int row = m0 + j + (half << 3);
      const float iv = sigmf(acc[0][j]);
      const float fv = sigmf(acc[1][j]);
      const float gv = fast_tanh(acc[2][j]);
      const float ov = sigmf(acc[3][j]);
      const float cv = fv * cl[row * HID + n] + iv * gv;
      cl[row * HID + n] = cv;
      hnew[ng][j] = ov * fast_tanh(cv);
    }
  }
#pragma unroll
  for (int ng = 0; ng < 4; ++ng) {
    const int n = ng * 16 + r;
#pragma unroll
    for (int j = 0; j < 8; ++j) {
      const int row = m0 + j + (half << 3);
      hl[row * HID + n] = hnew[ng][j];
      lin[row * HID + n] = hnew[ng][j];
    }
  }
}

__global__ __launch_bounds__(256) void decoder_k(
    const float* __restrict__ x,
    const float* __restrict__ Wih0, const float* __restrict__ WihL,
    const float* __restrict__ Whh, const float* __restrict__ biasL,
    const float* __restrict__ fcW, const float* __restrict__ fcB,
    const float* __restrict__ enc_h, const float* __restrict__ enc_c,
    const float* __restrict__ feat,
    float* hSt, float* cSt, float* __restrict__ out)
{
  __shared__ float lin[B_ * HID];
  __shared__ float inp[B_];

  const int lane = threadIdx.x & 31;
  const int wave = threadIdx.x >> 5;
  const int half = lane >> 4;
  const int r    = lane & 15;
  const int m0   = wave << 4;

  for (int l = 0; l < L_; ++l)
    for (int i = lane; i < 16 * HID; i += 32) {
      const int fidx = (m0 + (i >> 6)) * HID + (i & 63);
      const int idx = l * B_ * HID + fidx;
      hSt[idx] = enc_h[idx] + ALPHA_ * feat[fidx];
      cSt[idx] = enc_c[idx];
    }
  if (half == 0) inp[m0 + r] = x[(size_t)(m0 + r) * T_ + (T_ - 1)];

  for (int step = 0; step < PS_; ++step) {
    dec_cell<1>(Wih0, Whh, biasL, hSt, cSt, lin, inp, m0, half, r);
    for (int l = 1; l < L_; ++l) {
      dec_cell<HID>(WihL + (size_t)(l - 1) * G_ * HID,
                    Whh + (size_t)l * G_ * HID, biasL + l * G_,
                    hSt + (size_t)l * B_ * HID, cSt + (size_t)l * B_ * HID,
                    lin, inp, m0, half, r);
    }
    if (half == 0) {
      const int row = m0 + r;
      float s = fcB[0];
      for (int n = 0; n < HID; ++n) s += lin[row * HID + n] * fcW[n];
      out[row * PS_ + step] = s;
      inp[row] = s;
    }
  }
}

// ---------------------------------------------------------------------------
extern "C" void kernel_launch(void* const* d_in, const int* in_sizes, int n_in,
                              void* d_out, int out_size, void* d_ws, size_t ws_size,
                              hipStream_t stream) {
  (void)in_sizes; (void)n_in; (void)out_size; (void)ws_size;
  const float* x        = (const float*)d_in[0];
  const float* y        = (const float*)d_in[1];
  const float* enc_Wih0 = (const float*)d_in[2];
  const float* enc_Wih  = (const float*)d_in[3];
  const float* enc_Whh  = (const float*)d_in[4];
  const float* enc_b    = (const float*)d_in[5];
  const float* dec_Wih0 = (const float*)d_in[6];
  const float* dec_Wih  = (const float*)d_in[7];
  const float* dec_Whh  = (const float*)d_in[8];
  const float* dec_b    = (const float*)d_in[9];
  const float* fc_W     = (const float*)d_in[10];
  const float* fc_b     = (const float*)d_in[11];
  const float* conv0_W  = (const float*)d_in[12];
  const float* conv0_b  = (const float*)d_in[13];
  const float* convs_W  = (const float*)d_in[14];
  const float* convs_b  = (const float*)d_in[15];
  float* out = (float*)d_out;

  float* w     = (float*)d_ws;
  float* seqA  = w;                                // T*B*HID
  float* seqB  = seqA + (size_t)T_ * B_ * HID;
  float* convA = seqB + (size_t)T_ * B_ * HID;     // B*256*64 (NHWC)
  float* convB = convA + (size_t)B_ * 256 * 64;
  float* enc_h = convB + (size_t)B_ * 256 * 64;    // L*B*HID
  float* enc_c = enc_h + (size_t)L_ * B_ * HID;
  float* feat  = enc_c + (size_t)L_ * B_ * HID;    // B*HID
  float* dec_h = feat + (size_t)B_ * HID;          // L*B*HID
  float* dec_c = dec_h + (size_t)L_ * B_ * HID;
  float* Wr    = dec_c + (size_t)L_ * B_ * HID;    // 7*64*9*64 repacked conv W

  // ---- encoder ----
  lstm_layer_k<1, 1, T_><<<1, 256, 0, stream>>>(x, enc_Wih0, enc_Whh, enc_b,
                                                seqA, enc_h, enc_c);
  for (int l = 1; l < L_; ++l) {
    const float* sin = (l & 1) ? seqA : seqB;
    float* sout      = (l & 1) ? seqB : seqA;
    lstm_layer_k<HID, B_ * HID, HID><<<1, 256, 0, stream>>>(
        sin, enc_Wih + (size_t)(l - 1) * G_ * HID,
        enc_Whh + (size_t)l * G_ * HID, enc_b + l * G_,
        sout, enc_h + (size_t)l * B_ * HID, enc_c + (size_t)l * B_ * HID);
  }

  // ---- CNN branch ----
  repack_convW_k<<<(7 * 64 * 9 * 64) / 256, 256, 0, stream>>>(convs_W, Wr);
  conv0_pool_k<<<(B_ * 64 * 256) / 256, 256, 0, stream>>>(y, conv0_W, conv0_b,
                                                          convA);
  for (int i = 0; i < 7; ++i) {
    const float* ci = (i & 1) ? convB : convA;
    float* co       = (i & 1) ? convA : convB;
    conv3x3_relu_k<<<B_, 256, 0, stream>>>(ci, Wr + (size_t)i * 64 * 576,
                                           convs_b + i * 64, co);
  }
  feat_mean_k<<<B_, 64, 0, stream>>>(convB, feat);

  // ---- decoder ----
  decoder_k<<<1, 256, 0, stream>>>(x, dec_Wih0, dec_Wih, dec_Whh, dec_b,
                                   fc_W, fc_b, enc_h, enc_c, feat,
                                   dec_h, dec_c, out);
}